// AsyncGNN_54855322304747
// MI455X (gfx1250) — compile-verified
//
#include <hip/hip_runtime.h>

#define IN_CH      128
#define HID        128
#define OUT_CH     128
#define NUM_LEVELS 8

typedef __attribute__((ext_vector_type(2))) float v2f;
typedef __attribute__((ext_vector_type(8))) float v8f;

// ---------------------------------------------------------------------------
// Dense fp32 GEMM with bias:  out[m][n] = sum_k A[m][k] * W[n][k] + bias[n]
// A: [M,128] row-major, W: [128,128] row-major, out: [M,128].
// One wave computes a 16x16 tile via V_WMMA_F32_16X16X4_F32; 8 waves/block
// cover all 128 output columns of a 16-row stripe.
// A-frag layout: lane holds row (lane&15), K = k0 + (lane>>4)*2 + {0,1} -> v2f
// B-frag layout: lane holds col (lane&15) of W row-major, same K split -> v2f
// ---------------------------------------------------------------------------
__global__ __launch_bounds__(256)
void gemm128_bias(const float* __restrict__ A, const float* __restrict__ W,
                  const float* __restrict__ bias, float* __restrict__ out, int M)
{
    const int wave = threadIdx.x >> 5;
    const int lane = threadIdx.x & 31;
    const int half = lane >> 4;
    const int lr   = lane & 15;

    const int row0 = blockIdx.x * 16;
    if (row0 >= M) return;                       // uniform exit
    const int col0 = wave * 16;

    int arow = row0 + lr; if (arow >= M) arow = M - 1;   // clamp, no divergence
    const float* aptr = A + (size_t)arow * 128 + half * 2;
    const float* bptr = W + (size_t)(col0 + lr) * 128 + half * 2;

    v8f acc = {};
#pragma unroll 8
    for (int k0 = 0; k0 < 128; k0 += 4) {
        v2f a = *(const v2f*)(aptr + k0);
        v2f b = *(const v2f*)(bptr + k0);
        acc = __builtin_amdgcn_wmma_f32_16x16x4_f32(false, a, false, b,
                                                    (short)0, acc, false, false);
    }

    const float bv = bias[col0 + lr];
    // C/D layout: lanes 0-15 -> M = r, lanes 16-31 -> M = r + 8, N = lane&15.
    if (row0 + 16 <= M) {
        // Full tile (always the case when M % 16 == 0): branch-free stores
        // through one lane pointer with immediate row offsets (r*512B).
        float* optr = out + (size_t)(row0 + half * 8) * 128 + col0 + lr;
#pragma unroll
        for (int r = 0; r < 8; ++r) optr[(size_t)r * 128] = acc[r] + bv;
    } else {
#pragma unroll
        for (int r = 0; r < 8; ++r) {
            int row = row0 + r + half * 8;
            if (row < M) out[(size_t)row * 128 + col0 + lr] = acc[r] + bv;
        }
    }
}

// ---------------------------------------------------------------------------
// Bucket nodes by depth so level GEMMs touch only depth-k rows.
// ---------------------------------------------------------------------------
__global__ void build_lists(const int* __restrict__ depth, int* __restrict__ cnt,
                            int* __restrict__ list, int N)
{
    int i = blockIdx.x * blockDim.x + threadIdx.x;
    if (i >= N) return;
    int d = depth[i];
    if (d < 0) d = 0; if (d >= NUM_LEVELS) d = NUM_LEVELS - 1;
    int pos = atomicAdd(&cnt[d], 1);
    list[(size_t)d * N + pos] = i;
}

// ---------------------------------------------------------------------------
// Edge aggregation for one level: agg[v] += h[u] where depth[v]==level.
// One wave per edge; depth test is wave-uniform (whole-wave early exit);
// each lane handles 4 channels with fp32 global atomics.
// ---------------------------------------------------------------------------
__global__ __launch_bounds__(256)
void aggregate_level(const int* __restrict__ eu, const int* __restrict__ ev,
                     const int* __restrict__ depth,
                     const float* __restrict__ h, float* __restrict__ agg,
                     int level, int E)
{
    int e = blockIdx.x * 8 + (threadIdx.x >> 5);
    if (e >= E) return;
    int v = ev[e];
    if (depth[v] != level) return;               // wave-uniform
    int u = eu[e];
    int lane = threadIdx.x & 31;
    const float4 hv = *(const float4*)(h + (size_t)u * 128 + lane * 4);
    float* dst = agg + (size_t)v * 128 + lane * 4;
    atomicAdd(dst + 0, hv.x);
    atomicAdd(dst + 1, hv.y);
    atomicAdd(dst + 2, hv.z);
    atomicAdd(dst + 3, hv.w);
}

// ---------------------------------------------------------------------------
// Level update: for the cnt[level] nodes at depth==level,
//   h[node] = relu([h[node], agg[node]] @ W_linᵀ + b_lin)      (K = 256)
// Gathered 16x128 h/agg rows are staged into LDS (one read per block, and the
// __syncthreads() makes the in-place h update race-free across the 8 waves).
// ---------------------------------------------------------------------------
__global__ __launch_bounds__(256)
void level_update(const float* __restrict__ hin, const float* __restrict__ agg,
                  const float* __restrict__ W, const float* __restrict__ bias,
                  float* __restrict__ hout,
                  const int* __restrict__ list, const int* __restrict__ cnt,
                  int level, int N)
{
    __shared__ float sh[16 * 128];
    __shared__ float sa[16 * 128];

    const int c = cnt[level];
    const int row0 = blockIdx.x * 16;
    if (row0 >= c) return;                       // uniform exit

    const int wave = threadIdx.x >> 5;
    const int lane = threadIdx.x & 31;
    const int half = lane >> 4;
    const int lr   = lane & 15;
    const int col0 = wave * 16;
    const int* lst = list + (size_t)level * N;

    // Stage this block's 16 gathered rows of h and agg into LDS.
    {
        int t   = threadIdx.x;
        int row = t >> 4;
        int col = (t & 15) * 8;
        int ai  = row0 + row; if (ai >= c) ai = c - 1;
        int nd  = lst[ai];
        const float4* srch = (const float4*)(hin + (size_t)nd * 128 + col);
        const float4* srca = (const float4*)(agg + (size_t)nd * 128 + col);
        *(float4*)(&sh[row * 128 + col])     = srch[0];
        *(float4*)(&sh[row * 128 + col + 4]) = srch[1];
        *(float4*)(&sa[row * 128 + col])     = srca[0];
        *(float4*)(&sa[row * 128 + col + 4]) = srca[1];
    }
    __syncthreads();

    const float* bptr = W + (size_t)(col0 + lr) * 256 + half * 2;

    v8f acc = {};
#pragma unroll 8
    for (int k0 = 0; k0 < 128; k0 += 4) {        // h half of K
        v2f a = *(const v2f*)(&sh[lr * 128 + k0 + half * 2]);
        v2f b = *(const v2f*)(bptr + k0);
        acc = __builtin_amdgcn_wmma_f32_16x16x4_f32(false, a, false, b,
                                                    (short)0, acc, false, false);
    }
#pragma unroll 8
    for (int k0 = 0; k0 < 128; k0 += 4) {        // agg half of K
        v2f a = *(const v2f*)(&sa[lr * 128 + k0 + half * 2]);
        v2f b = *(const v2f*)(bptr + 128 + k0);
        acc = __builtin_amdgcn_wmma_f32_16x16x4_f32(false, a, false, b,
                                                    (short)0, acc, false, false);
    }

    const float bv = bias[col0 + lr];
    if (row0 + 16 <= c) {
        // Full tile: no exec masking; gather the 8 destination node ids and
        // issue branch-free scattered stores.
        int nd[8];
#pragma unroll
        for (int r = 0; r < 8; ++r) nd[r] = lst[row0 + r + half * 8];
#pragma unroll
        for (int r = 0; r < 8; ++r)
            hout[(size_t)nd[r] * 128 + col0 + lr] = fmaxf(acc[r] + bv, 0.0f);
    } else {
#pragma unroll
        for (int r = 0; r < 8; ++r) {
            int ridx = row0 + r + half * 8;
            if (ridx < c) {
                int nd = lst[ridx];
                hout[(size_t)nd * 128 + col0 + lr] = fmaxf(acc[r] + bv, 0.0f);
            }
        }
    }
}

// ---------------------------------------------------------------------------
// Orchestration
// ---------------------------------------------------------------------------
extern "C" void kernel_launch(void* const* d_in, const int* in_sizes, int n_in,
                              void* d_out, int out_size, void* d_ws, size_t ws_size,
                              hipStream_t stream)
{
    const float* x     = (const float*)d_in[0];
    const int*   edges = (const int*)  d_in[1];   // [2,E]: u rows then v rows
    const int*   depth = (const int*)  d_in[2];
    const float* W_in  = (const float*)d_in[3];
    const float* b_in  = (const float*)d_in[4];
    const float* W_lin = (const float*)d_in[5];   // [128,256]
    const float* b_lin = (const float*)d_in[6];
    const float* W_out = (const float*)d_in[7];
    const float* b_out = (const float*)d_in[8];

    const int N = in_sizes[2];
    const int E = in_sizes[1] / 2;
    const int* eu = edges;
    const int* ev = edges + E;

    // Workspace layout: h | agg | lists | counts  (~106 MB for N=100k)
    char*  ws   = (char*)d_ws;
    float* h    = (float*)ws;
    float* agg  = (float*)(ws + (size_t)N * HID * sizeof(float));
    int*   list = (int*)  (ws + 2 * (size_t)N * HID * sizeof(float));
    int*   cnt  = (int*)  (ws + 2 * (size_t)N * HID * sizeof(float)
                              + (size_t)NUM_LEVELS * N * sizeof(int));

    // agg is written/read at exactly one level per node -> zero once per call.
    hipMemsetAsync(agg, 0, (size_t)N * HID * sizeof(float), stream);
    hipMemsetAsync(cnt, 0, NUM_LEVELS * sizeof(int), stream);

    build_lists<<<(N + 255) / 256, 256, 0, stream>>>(depth, cnt, list, N);

    const int mtiles  = (N + 15) / 16;
    const int eblocks = (E + 7) / 8;

    // h = x @ W_inᵀ + b_in
    gemm128_bias<<<mtiles, 256, 0, stream>>>(x, W_in, b_in, h, N);

    for (int k = 0; k < NUM_LEVELS; ++k) {
        aggregate_level<<<eblocks, 256, 0, stream>>>(eu, ev, depth, h, agg, k, E);
        level_update<<<mtiles, 256, 0, stream>>>(h, agg, W_lin, b_lin, h,
                                                 list, cnt, k, N);
    }

    // z = h @ W_outᵀ + b_out
    gemm128_bias<<<mtiles, 256, 0, stream>>>(h, W_out, b_out, (float*)d_out, N);
}